// YksModel_BERT_FC1_78563541779031
// MI455X (gfx1250) — compile-verified
//
#include <hip/hip_runtime.h>
#include <cmath>

typedef __attribute__((ext_vector_type(2))) float v2f;
typedef __attribute__((ext_vector_type(4))) float v4f;
typedef __attribute__((ext_vector_type(8))) float v8f;

#define B_  64
#define S_  512
#define H_  768
#define U_  16
#define C_  5
#define SC_ 8            // chunks along S
#define TOK_ (S_ / SC_)  // 64 tokens per chunk

// ---------------- kernel 0: zero atomic accumulators ----------------
__global__ void zero_ws_k(float* __restrict__ p, int n) {
    int i = blockIdx.x * 256 + threadIdx.x;
    if (i < n) p[i] = 0.0f;
}

// ---------------- kernel 1: ragged per-(batch,utterance) partial sums ----------------
// grid = B_*SC_ blocks of 192 threads; thread owns 4 contiguous columns (192*4 = 768).
// segment_ids are sorted per row, so a chunk is a handful of contiguous runs:
// accumulate each run in registers, flush at run boundaries via f32 global atomics.
__global__ __launch_bounds__(192) void seg_partial_k(const float* __restrict__ hidden,
                                                     const int*   __restrict__ seg,
                                                     float* __restrict__ gsum,
                                                     float* __restrict__ gcnt) {
    __shared__ int lsid[TOK_];
    __shared__ int lcnt[U_];
    const int tid = threadIdx.x;
    const int b   = blockIdx.x / SC_;
    const int t0  = (blockIdx.x % SC_) * TOK_;

    if (tid < U_) lcnt[tid] = 0;
    __syncthreads();
    if (tid < TOK_) {
        int s = seg[b * S_ + t0 + tid];
        lsid[tid] = s;
        atomicAdd(&lcnt[s], 1);
    }
    __syncthreads();

    const float* base = hidden + (size_t)(b * S_ + t0) * H_ + tid * 4;

    v4f acc; acc.x = 0.f; acc.y = 0.f; acc.z = 0.f; acc.w = 0.f;
    int cur = lsid[0];
    for (int t = 0; t < TOK_; ++t) {
        const int s = lsid[t];            // block-uniform -> scalar branch below
        if (s != cur) {
            float* g = gsum + (size_t)(b * U_ + cur) * H_ + tid * 4;
            unsafeAtomicAdd(g + 0, acc.x);
            unsafeAtomicAdd(g + 1, acc.y);
            unsafeAtomicAdd(g + 2, acc.z);
            unsafeAtomicAdd(g + 3, acc.w);
            acc.x = 0.f; acc.y = 0.f; acc.z = 0.f; acc.w = 0.f;
            cur = s;
        }
        // streamed exactly once -> non-temporal b128 load
        v4f v = __builtin_nontemporal_load((const v4f*)(base + (size_t)t * H_));
        acc += v;
    }
    {   // final run flush
        float* g = gsum + (size_t)(b * U_ + cur) * H_ + tid * 4;
        unsafeAtomicAdd(g + 0, acc.x);
        unsafeAtomicAdd(g + 1, acc.y);
        unsafeAtomicAdd(g + 2, acc.z);
        unsafeAtomicAdd(g + 3, acc.w);
    }
    if (tid < U_ && lcnt[tid] > 0)
        unsafeAtomicAdd(&gcnt[b * U_ + tid], (float)lcnt[tid]);
}

// ---------------- kernel 2: means = sums / max(count,1) ----------------
__global__ void mean_k(const float* __restrict__ gsum, const float* __restrict__ gcnt,
                       float* __restrict__ means) {
    int i = blockIdx.x * 256 + threadIdx.x;
    if (i >= B_ * U_ * H_) return;
    int   r = i / H_;
    float c = gcnt[r];
    c = c > 1.0f ? c : 1.0f;
    means[i] = gsum[i] / c;
}

// ---------------- kernel 3: fp32 WMMA GEMM (means @ W_h) + bias + SELU ----------------
// One 16x16 output tile per wave; K-loop of v_wmma_f32_16x16x4_f32.
// Block = 8 waves sharing the same 16 A-rows (mt), covering 8 consecutive nt.
__global__ __launch_bounds__(256) void gemm1_selu_k(const float* __restrict__ A,   // [1024,768]
                                                    const float* __restrict__ Wh,  // [768,768]
                                                    const float* __restrict__ bh,  // [768]
                                                    float* __restrict__ Hout) {    // [1024,768]
    const int wave = threadIdx.x >> 5;
    const int lane = threadIdx.x & 31;
    const int half = lane >> 4;   // splits K across half-waves (32-bit A/B layout)
    const int l15  = lane & 15;

    const int mt   = blockIdx.x / 6;               // 64 M-tiles
    const int nt   = (blockIdx.x % 6) * 8 + wave;  // 48 N-tiles
    const int row0 = mt * 16;
    const int col0 = nt * 16;

    // A fragment: lane l (M = l&15) holds K = {2*half, 2*half+1} -> contiguous float2
    const float* arow = A + (size_t)(row0 + l15) * H_ + 2 * half;
    // B fragment: lane l (N = l&15) holds K = {2*half, 2*half+1} (stride H_ apart)
    const float* bcol = Wh + (size_t)(2 * half) * H_ + col0 + l15;

    v8f c;
    #pragma unroll
    for (int j = 0; j < 8; ++j) c[j] = 0.0f;

    #pragma unroll 4
    for (int k = 0; k < H_; k += 4) {
        v2f a = *(const v2f*)(arow + k);
        v2f b;
        b.x = bcol[(size_t)k * H_];
        b.y = bcol[(size_t)(k + 1) * H_];
        c = __builtin_amdgcn_wmma_f32_16x16x4_f32(
                /*neg_a=*/false, a, /*neg_b=*/false, b,
                /*c_mod=*/(short)0, c, /*reuse_a=*/false, /*reuse_b=*/false);
    }

    const float bias  = bh[col0 + l15];
    const float scale = 1.0507009873554805f;
    const float alpha = 1.6732632423543772f;
    #pragma unroll
    for (int j = 0; j < 8; ++j) {
        // C/D layout: VGPR j, lanes 0-15 -> M=j, lanes 16-31 -> M=j+8, N = l15
        const int row = row0 + j + 8 * half;
        float x = c[j] + bias;
        float y = x > 0.0f ? scale * x : scale * alpha * (expf(x) - 1.0f);
        Hout[(size_t)row * H_ + col0 + l15] = y;
    }
}

// ---------------- kernel 4: head GEMM (h @ W_o + b_o) + softmax ----------------
// One wave per output row (C=5 is too skinny for WMMA tiles); W_o staged in LDS.
__global__ __launch_bounds__(256) void head_softmax_k(const float* __restrict__ Hin, // [1024,768]
                                                      const float* __restrict__ Wo,  // [768,5]
                                                      const float* __restrict__ bo,  // [5]
                                                      float* __restrict__ out) {     // [1024,5]
    __shared__ float lwo[H_ * C_];  // 15 KB
    const int tid = threadIdx.x;
    for (int i = tid; i < H_ * C_; i += 256) lwo[i] = Wo[i];
    __syncthreads();

    const int wave = tid >> 5;
    const int lane = tid & 31;
    const int row  = blockIdx.x * 8 + wave;

    const float* hr = Hin + (size_t)row * H_;
    float acc[C_] = {0.f, 0.f, 0.f, 0.f, 0.f};
    for (int k = lane; k < H_; k += 32) {
        const float hv = hr[k];
        #pragma unroll
        for (int c = 0; c < C_; ++c) acc[c] += hv * lwo[k * C_ + c];
    }
    #pragma unroll
    for (int m = 16; m >= 1; m >>= 1) {
        #pragma unroll
        for (int c = 0; c < C_; ++c) acc[c] += __shfl_xor(acc[c], m, 32);
    }
    if (lane == 0) {
        float lg[C_];
        float mx = -3.4e38f;
        #pragma unroll
        for (int c = 0; c < C_; ++c) { lg[c] = acc[c] + bo[c]; mx = fmaxf(mx, lg[c]); }
        float sum = 0.0f;
        #pragma unroll
        for (int c = 0; c < C_; ++c) { lg[c] = expf(lg[c] - mx); sum += lg[c]; }
        const float inv = 1.0f / sum;
        #pragma unroll
        for (int c = 0; c < C_; ++c) out[row * C_ + c] = lg[c] * inv;
    }
}

// ---------------- launcher ----------------
extern "C" void kernel_launch(void* const* d_in, const int* in_sizes, int n_in,
                              void* d_out, int out_size, void* d_ws, size_t ws_size,
                              hipStream_t stream) {
    const float* hidden = (const float*)d_in[0];  // [64,512,768] f32
    const int*   seg    = (const int*)  d_in[1];  // [64,512] i32
    // d_in[2] = n_utter scalar (16), compile-time constant here
    const float* Wh     = (const float*)d_in[3];  // [768,768]
    const float* bh     = (const float*)d_in[4];  // [768]
    const float* Wo     = (const float*)d_in[5];  // [768,5]
    const float* bo     = (const float*)d_in[6];  // [5]
    float* out = (float*)d_out;                   // [1024,5] f32

    float* ws    = (float*)d_ws;
    float* gsum  = ws;                               // 1024*768
    float* gcnt  = gsum + (size_t)B_ * U_ * H_;      // 1024
    float* means = gcnt + B_ * U_;                   // 1024*768
    float* hbuf  = means + (size_t)B_ * U_ * H_;     // 1024*768

    const int nz = B_ * U_ * H_ + B_ * U_;           // gsum + gcnt (contiguous)
    zero_ws_k<<<(nz + 255) / 256, 256, 0, stream>>>(gsum, nz);

    seg_partial_k<<<B_ * SC_, 192, 0, stream>>>(hidden, seg, gsum, gcnt);

    const int nm = B_ * U_ * H_;
    mean_k<<<(nm + 255) / 256, 256, 0, stream>>>(gsum, gcnt, means);

    // 64 M-tiles x 48 N-tiles, 8 waves (tiles) per block -> 384 blocks
    gemm1_selu_k<<<(B_ * U_ / 16) * ((H_ / 16) / 8), 256, 0, stream>>>(means, Wh, bh, hbuf);

    head_softmax_k<<<B_ * U_ / 8, 256, 0, stream>>>(hbuf, Wo, bo, out);
}